// PEG_Shift_171798692034
// MI455X (gfx1250) — compile-verified
//
#include <hip/hip_runtime.h>

// ---------------------------------------------------------------------------
// 3x3 depthwise conv (pad=1, stride=1) with deepshift-style quantization.
// One workgroup per (n,c) 56x56 plane, staged via CDNA5 async load-to-LDS.
// Quad-per-thread compute to minimize LDS bytes per output.
// ---------------------------------------------------------------------------

#define DIMC      384
#define HW        56
#define PLANE     (HW * HW)        // 3136 elements per (n,c) plane
#define LDS_STRIDE 64              // floats; keeps 16B chunks 16B aligned
#define LDS_ROWS   58              // 56 + top/bottom halo
#define PADCOL     4               // interior starts at column 4 (halo at 3 / 60)
#define NCHUNK    (PLANE / 4)      // 784 b128 chunks per plane (14 per row)

#ifndef __has_builtin
#define __has_builtin(x) 0
#endif

typedef int v4i __attribute__((ext_vector_type(4)));
typedef __attribute__((address_space(1))) v4i gv4i;   // global int4
typedef __attribute__((address_space(3))) v4i lv4i;   // LDS int4

__device__ __forceinline__ void async_load16(const float* g, float* lds_base,
                                             int lds_byte_off) {
  // Low 32 bits of a generic shared-memory address == LDS byte offset
  // (ISA aperture rule).
  const unsigned lds_addr =
      (unsigned)(unsigned long long)lds_base + (unsigned)lds_byte_off;
#if __has_builtin(__builtin_amdgcn_global_load_async_to_lds_b128)
  __builtin_amdgcn_global_load_async_to_lds_b128(
      (gv4i*)(unsigned long long)g,   // AS1 int4*
      (lv4i*)lds_addr,                // AS3 int4* (32-bit int -> 32-bit ptr)
      0, 0);
#else
  asm volatile("global_load_async_to_lds_b128 %0, %1, off"
               :
               : "v"(lds_addr), "v"(g)
               : "memory");
#endif
}

__device__ __forceinline__ void wait_async0() {
#if __has_builtin(__builtin_amdgcn_s_wait_asynccnt)
  __builtin_amdgcn_s_wait_asynccnt(0);
#else
  asm volatile("s_wait_asynccnt 0" ::: "memory");
#endif
}

__global__ __launch_bounds__(256) void peg_shift_dwconv3x3(
    const float* __restrict__ x, const float* __restrict__ w,
    float* __restrict__ out) {
  __shared__ float lds[LDS_ROWS * LDS_STRIDE];  // 14848 bytes

  const int tid   = threadIdx.x;
  const int plane = blockIdx.x;                 // n * DIMC + c
  const int c     = plane % DIMC;
  const float* xp = x + (size_t)plane * PLANE;
  float*       op = out + (size_t)plane * PLANE;

  // --- zero the halo border (disjoint from async-written interior cells) ----
  if (tid < 228) {
    int idx;
    if (tid < 58)       idx = 3 + tid;                               // top row
    else if (tid < 116) idx = 57 * LDS_STRIDE + 3 + (tid - 58);      // bottom
    else if (tid < 172) idx = (tid - 116 + 1) * LDS_STRIDE + 3;      // left
    else                idx = (tid - 172 + 1) * LDS_STRIDE + 60;     // right
    lds[idx] = 0.0f;
  }

  // --- async-stage the 56x56 plane into the padded LDS tile -----------------
  for (int chunk = tid; chunk < NCHUNK; chunk += 256) {
    const int row  = chunk / 14;          // 14 x b128 per image row
    const int col4 = (chunk % 14) * 4;    // float column within the row
    const int loff = ((row + 1) * LDS_STRIDE + PADCOL + col4) * 4;
    async_load16(xp + chunk * 4, lds, loff);
  }

  // --- quantize the 9 per-channel weights while loads are in flight ---------
  // wq = sign(w) * 2^clip(rint(log2(max(|w|, 2^-20))), -14, 0)
  float wq[9];
#pragma unroll
  for (int j = 0; j < 9; ++j) {
    const float wv = w[c * 9 + j];
    const float aw = fabsf(wv);
    float sh = rintf(log2f(fmaxf(aw, 9.5367431640625e-7f)));  // 2^-20 floor
    sh = fminf(fmaxf(sh, -14.0f), 0.0f);
    const float sg = (wv > 0.0f) ? 1.0f : ((wv < 0.0f) ? -1.0f : 0.0f);
    wq[j] = sg * exp2f(sh);
  }

  wait_async0();
  __syncthreads();

  // --- in-place fixed-point quantization (16 frac bits), b128 granularity ---
  const float SC = 65536.0f, ISC = 1.0f / 65536.0f;
  for (int q = tid; q < NCHUNK; q += 256) {
    const int row  = q / 14;
    const int col4 = (q % 14) * 4;
    float4* p = (float4*)&lds[(row + 1) * LDS_STRIDE + PADCOL + col4];
    float4 v = *p;
    v.x = rintf(v.x * SC) * ISC;
    v.y = rintf(v.y * SC) * ISC;
    v.z = rintf(v.z * SC) * ISC;
    v.w = rintf(v.w * SC) * ISC;
    *p = v;
  }
  __syncthreads();

  // --- 4 horizontal outputs per thread from the zero-haloed tile ------------
  // For output quad x = col4..col4+3 of row y, each input row contributes the
  // 6 floats at padded columns col4+3 .. col4+8: loaded as b32 + b128 + b32.
  for (int q = tid; q < NCHUNK; q += 256) {
    const int y    = q / 14;
    const int col4 = (q % 14) * 4;
    float a0 = 0.f, a1 = 0.f, a2 = 0.f, a3 = 0.f;
#pragma unroll
    for (int ky = 0; ky < 3; ++ky) {
      const int base = (y + ky) * LDS_STRIDE + col4;   // haloed row y-1+ky
      const float  lo = lds[base + 3];
      const float4 m  = *(const float4*)&lds[base + 4];  // aligned b128
      const float  hi = lds[base + 8];
      const float w0 = wq[ky * 3 + 0], w1 = wq[ky * 3 + 1], w2 = wq[ky * 3 + 2];
      a0 += lo  * w0 + m.x * w1 + m.y * w2;
      a1 += m.x * w0 + m.y * w1 + m.z * w2;
      a2 += m.y * w0 + m.z * w1 + m.w * w2;
      a3 += m.z * w0 + m.w * w1 + hi  * w2;
    }
    *(float4*)(op + q * 4) = make_float4(a0, a1, a2, a3);  // coalesced b128
  }
}

extern "C" void kernel_launch(void* const* d_in, const int* in_sizes, int n_in,
                              void* d_out, int out_size, void* d_ws, size_t ws_size,
                              hipStream_t stream) {
  const float* x = (const float*)d_in[0];   // (64, 384, 56, 56) fp32
  const float* w = (const float*)d_in[1];   // (384, 1, 3, 3)    fp32
  float* out = (float*)d_out;               // (64, 384, 56, 56) fp32

  const int nplanes = in_sizes[0] / PLANE;  // 64 * 384 = 24576
  peg_shift_dwconv3x3<<<nplanes, 256, 0, stream>>>(x, w, out);
}